// DynamicGAT_3607772529310
// MI455X (gfx1250) — compile-verified
//
#include <hip/hip_runtime.h>
#include <hip/hip_bf16.h>
#include <math.h>

#define N_NODESC 50000
#define N_EDGESC 400000
#define E_TOT    (N_EDGESC + N_NODESC)   // self loops appended
#define DIMC     64
#define HEADSC   8
#define HDC      (HEADSC * DIMC)         // 512
#define NUM_G    64
#define OUT_DIMC 32
#define NEG_SL   0.2f
#define LN_EPS   1e-5f

typedef __attribute__((ext_vector_type(2))) float v2f;
typedef __attribute__((ext_vector_type(8))) float v8f;

// order-preserving float <-> uint transform for atomic float-max via atomicMax(u32)
__device__ __forceinline__ unsigned f2ord(float f) {
  unsigned u = __float_as_uint(f);
  return (u & 0x80000000u) ? ~u : (u | 0x80000000u);
}
__device__ __forceinline__ float ord2f(unsigned u) {
  return (u & 0x80000000u) ? __uint_as_float(u & 0x7fffffffu) : __uint_as_float(~u);
}

__global__ void k_zero(float* __restrict__ p, int n) {
  int i = blockIdx.x * blockDim.x + threadIdx.x;
  if (i < n) p[i] = 0.0f;
}

__global__ void k_count(const long long* __restrict__ batch, float* __restrict__ gcnt, int n) {
  int i = blockIdx.x * blockDim.x + threadIdx.x;
  if (i < n) atomicAdd(&gcnt[(int)batch[i]], 1.0f);
}

// H[N,512] = X[N,64] @ W[64,512] using V_WMMA_F32_16X16X4_F32.
// One block = one 16-row tile (N = 3125 * 16 exactly, no tail), 32 waves cover
// the 32 column tiles. A tile staged through LDS (stride 65 avoids bank conflicts).
__global__ __launch_bounds__(1024) void k_gemm(const float* __restrict__ X,
                                               const float* __restrict__ W,
                                               float* __restrict__ H) {
  __shared__ float As[16 * 65];
  const int row0 = blockIdx.x * 16;
  const int t = threadIdx.x;
  {
    int r = t >> 6, c = t & 63;                 // 1024 threads load 16x64 tile
    As[r * 65 + c] = X[(size_t)(row0 + r) * DIMC + c];
  }
  __syncthreads();

  const int wave = t >> 5;                      // col tile 0..31
  const int lane = t & 31;
  const int half = lane >> 4;                   // K-half select
  const int l = lane & 15;
  const int col0 = wave * 16;

  v8f c = {};
  #pragma unroll
  for (int k0 = 0; k0 < DIMC; k0 += 4) {
    const int k = k0 + (half ? 2 : 0);          // ISA 32-bit A 16x4: lo lanes K=0,1; hi lanes K=2,3
    v2f a, b;
    a.x = As[l * 65 + k];
    a.y = As[l * 65 + k + 1];
    b.x = W[(size_t)k * HDC + col0 + l];        // B 4x16: row K striped across lanes (N=l)
    b.y = W[(size_t)(k + 1) * HDC + col0 + l];
    c = __builtin_amdgcn_wmma_f32_16x16x4_f32(false, a, false, b, (short)0, c, false, false);
  }
  #pragma unroll
  for (int r = 0; r < 8; ++r) {                 // C/D: vgpr r -> M = r + 8*half, N = l
    const int row = row0 + r + (half ? 8 : 0);
    H[(size_t)row * HDC + col0 + l] = c[r];
  }
}

// al_s[n,h] = sum_d H[n,h,d] * a_src[h,d]  (and same for a_dst). Wave per node.
__global__ void k_attn(const float* __restrict__ H, const float* __restrict__ a_src,
                       const float* __restrict__ a_dst, float* __restrict__ als,
                       float* __restrict__ ald, int n) {
  int wave = (blockIdx.x * blockDim.x + threadIdx.x) >> 5;
  int lane = threadIdx.x & 31;
  if (wave >= n) return;
  const float* hr = H + (size_t)wave * HDC;
  #pragma unroll
  for (int hd = 0; hd < HEADSC; ++hd) {
    float v0 = hr[hd * DIMC + lane];
    float v1 = hr[hd * DIMC + lane + 32];
    float s = v0 * a_src[hd * DIMC + lane] + v1 * a_src[hd * DIMC + lane + 32];
    float d = v0 * a_dst[hd * DIMC + lane] + v1 * a_dst[hd * DIMC + lane + 32];
    #pragma unroll
    for (int off = 16; off; off >>= 1) {
      s += __shfl_xor(s, off, 32);
      d += __shfl_xor(d, off, 32);
    }
    if (lane == 0) { als[wave * HEADSC + hd] = s; ald[wave * HEADSC + hd] = d; }
  }
}

__device__ __forceinline__ void edge_sd(const long long* __restrict__ ei, int e, int& s, int& d) {
  if (e < N_EDGESC) { s = (int)ei[e]; d = (int)ei[N_EDGESC + e]; }
  else              { s = d = e - N_EDGESC; }   // appended self loops
}

// pass A: logits = leaky_relu(al_s[src] + al_d[dst]); segment max per (dst, head)
__global__ void k_edge_logits(const long long* __restrict__ ei, const float* __restrict__ als,
                              const float* __restrict__ ald, float* __restrict__ logits,
                              unsigned* __restrict__ smax) {
  int idx = blockIdx.x * blockDim.x + threadIdx.x;
  if (idx >= E_TOT * HEADSC) return;
  int e = idx >> 3, hd = idx & 7, s, d;
  edge_sd(ei, e, s, d);
  float z = als[s * HEADSC + hd] + ald[d * HEADSC + hd];
  z = (z > 0.0f) ? z : NEG_SL * z;
  logits[idx] = z;
  atomicMax(&smax[d * HEADSC + hd], f2ord(z));
}

// pass B: e = exp(logit - max); segment sum per (dst, head)
__global__ void k_edge_expsum(const long long* __restrict__ ei, float* __restrict__ logits,
                              const unsigned* __restrict__ smax, float* __restrict__ ssum) {
  int idx = blockIdx.x * blockDim.x + threadIdx.x;
  if (idx >= E_TOT * HEADSC) return;
  int e = idx >> 3, hd = idx & 7, s, d;
  edge_sd(ei, e, s, d);
  float ev = expf(logits[idx] - ord2f(smax[d * HEADSC + hd]));
  logits[idx] = ev;
  atomicAdd(&ssum[d * HEADSC + hd], ev);
}

// pass C: out[dst,:64] += (1/H) * sum_h alpha(e,h) * H[src,h,:]. Wave per edge.
__global__ void k_aggregate(const long long* __restrict__ ei, const float* __restrict__ logits,
                            const float* __restrict__ ssum, const float* __restrict__ H,
                            float* __restrict__ out) {
  int wave = (blockIdx.x * blockDim.x + threadIdx.x) >> 5;
  int lane = threadIdx.x & 31;
  if (wave >= E_TOT) return;
  int s, d;
  edge_sd(ei, wave, s, d);
  float wv = 0.0f;
  if (lane < HEADSC)
    wv = logits[(size_t)wave * HEADSC + lane] /
         (ssum[d * HEADSC + lane] + 1e-16f) * (1.0f / HEADSC);
  const float* hs = H + (size_t)s * HDC;
  float a0 = 0.0f, a1 = 0.0f;
  #pragma unroll
  for (int hd = 0; hd < HEADSC; ++hd) {
    float w = __shfl(wv, hd, 32);
    a0 += w * hs[hd * DIMC + lane];
    a1 += w * hs[hd * DIMC + lane + 32];
  }
  atomicAdd(&out[(size_t)d * DIMC + lane], a0);
  atomicAdd(&out[(size_t)d * DIMC + lane + 32], a1);
}

// y = xin + agg + b; accumulate per-graph sum(y), sum(y^2). Wave per node.
__global__ void k_residual_stats(const float* __restrict__ xin, const float* __restrict__ agg,
                                 const float* __restrict__ b, const long long* __restrict__ batch,
                                 float* __restrict__ y, float* __restrict__ gsum,
                                 float* __restrict__ gsq, int n) {
  int wave = (blockIdx.x * blockDim.x + threadIdx.x) >> 5;
  int lane = threadIdx.x & 31;
  if (wave >= n) return;
  size_t base = (size_t)wave * DIMC;
  float v0 = xin[base + lane]      + agg[base + lane]      + b[lane];
  float v1 = xin[base + lane + 32] + agg[base + lane + 32] + b[lane + 32];
  y[base + lane] = v0;
  y[base + lane + 32] = v1;
  float s = v0 + v1, s2 = v0 * v0 + v1 * v1;
  #pragma unroll
  for (int off = 16; off; off >>= 1) {
    s  += __shfl_xor(s, off, 32);
    s2 += __shfl_xor(s2, off, 32);
  }
  if (lane == 0) {
    int g = (int)batch[wave];
    atomicAdd(&gsum[g], s);
    atomicAdd(&gsq[g], s2);
  }
}

// graph LayerNorm (mean/var over all nodes*channels of each graph) + ReLU, in place
__global__ void k_norm_relu(float* __restrict__ y, const long long* __restrict__ batch,
                            const float* __restrict__ gsum, const float* __restrict__ gsq,
                            const float* __restrict__ gcnt, const float* __restrict__ w,
                            const float* __restrict__ b, int total) {
  int i = blockIdx.x * blockDim.x + threadIdx.x;
  if (i >= total) return;
  int nidx = i >> 6, d = i & 63;
  int g = (int)batch[nidx];
  float norm = fmaxf(gcnt[g], 1.0f) * (float)DIMC;
  float mean = gsum[g] / norm;
  float var  = gsq[g] / norm - mean * mean;
  float v = (y[i] - mean) * rsqrtf(var + LN_EPS) * w[d] + b[d];
  y[i] = fmaxf(v, 0.0f);
}

__global__ void k_pool(const float* __restrict__ y, const long long* __restrict__ batch,
                       float* __restrict__ pooled, int total) {
  int i = blockIdx.x * blockDim.x + threadIdx.x;
  if (i >= total) return;
  int nidx = i >> 6, d = i & 63;
  atomicAdd(&pooled[(int)batch[nidx] * DIMC + d], y[i]);
}

__global__ void k_head(const float* __restrict__ pooled, const float* __restrict__ gcnt,
                       const float* __restrict__ Wh, const float* __restrict__ bh,
                       float* __restrict__ out) {
  int i = blockIdx.x * blockDim.x + threadIdx.x;
  if (i >= NUM_G * OUT_DIMC) return;
  int g = i / OUT_DIMC, o = i % OUT_DIMC;
  float inv = 1.0f / fmaxf(gcnt[g], 1.0f);
  float acc = bh[o];
  #pragma unroll 8
  for (int d = 0; d < DIMC; ++d)
    acc += pooled[g * DIMC + d] * inv * Wh[d * OUT_DIMC + o];
  out[i] = fmaxf(acc, 0.0f);
}

static inline int cdiv(long long a, long long b) { return (int)((a + b - 1) / b); }

extern "C" void kernel_launch(void* const* d_in, const int* in_sizes, int n_in,
                              void* d_out, int out_size, void* d_ws, size_t ws_size,
                              hipStream_t stream) {
  const float*     x     = (const float*)d_in[0];
  const long long* ei    = (const long long*)d_in[1];
  const long long* batch = (const long long*)d_in[2];
  const float* W1  = (const float*)d_in[3];
  const float* as1 = (const float*)d_in[4];
  const float* ad1 = (const float*)d_in[5];
  const float* b1  = (const float*)d_in[6];
  const float* lw1 = (const float*)d_in[7];
  const float* lb1 = (const float*)d_in[8];
  const float* W2  = (const float*)d_in[9];
  const float* as2 = (const float*)d_in[10];
  const float* ad2 = (const float*)d_in[11];
  const float* b2  = (const float*)d_in[12];
  const float* lw2 = (const float*)d_in[13];
  const float* lb2 = (const float*)d_in[14];
  const float* Wh  = (const float*)d_in[15];
  const float* bh  = (const float*)d_in[16];
  float* out = (float*)d_out;

  // workspace layout (floats)
  float* ws = (float*)d_ws;
  float*    h      = ws;                                    // N*512
  float*    als    = h + (size_t)N_NODESC * HDC;            // N*8
  float*    ald    = als + (size_t)N_NODESC * HEADSC;       // N*8
  unsigned* smax   = (unsigned*)(ald + (size_t)N_NODESC * HEADSC); // N*8
  float*    ssum   = (float*)smax + (size_t)N_NODESC * HEADSC;     // N*8
  float*    logits = ssum + (size_t)N_NODESC * HEADSC;      // E_TOT*8
  float*    bufin  = logits + (size_t)E_TOT * HEADSC;       // N*64
  float*    bufout = bufin + (size_t)N_NODESC * DIMC;       // N*64
  float*    gsum   = bufout + (size_t)N_NODESC * DIMC;      // 64
  float*    gsq    = gsum + NUM_G;                          // 64
  float*    gcnt   = gsq + NUM_G;                           // 64
  float*    pooled = gcnt + NUM_G;                          // 64*64

  const int B = 256;

  // per-graph node counts + pooled init (gcnt and pooled are contiguous)
  k_zero<<<cdiv(NUM_G + NUM_G * DIMC, B), B, 0, stream>>>(gcnt, NUM_G + NUM_G * DIMC);
  k_count<<<cdiv(N_NODESC, B), B, 0, stream>>>(batch, gcnt, N_NODESC);

  auto layer = [&](const float* xin, const float* W, const float* as_, const float* ad_,
                   const float* bb, const float* lw, const float* lb) {
    k_zero<<<cdiv(N_NODESC * HEADSC, B), B, 0, stream>>>((float*)smax, N_NODESC * HEADSC);
    k_zero<<<cdiv(N_NODESC * HEADSC, B), B, 0, stream>>>(ssum, N_NODESC * HEADSC);
    k_zero<<<cdiv(N_NODESC * DIMC, B), B, 0, stream>>>(bufout, N_NODESC * DIMC);
    k_zero<<<1, B, 0, stream>>>(gsum, 2 * NUM_G);           // gsum + gsq contiguous

    k_gemm<<<N_NODESC / 16, 1024, 0, stream>>>(xin, W, h);
    k_attn<<<cdiv((long long)N_NODESC * 32, B), B, 0, stream>>>(h, as_, ad_, als, ald, N_NODESC);
    k_edge_logits<<<cdiv((long long)E_TOT * HEADSC, B), B, 0, stream>>>(ei, als, ald, logits, smax);
    k_edge_expsum<<<cdiv((long long)E_TOT * HEADSC, B), B, 0, stream>>>(ei, logits, smax, ssum);
    k_aggregate<<<cdiv((long long)E_TOT * 32, B), B, 0, stream>>>(ei, logits, ssum, h, bufout);
    k_residual_stats<<<cdiv((long long)N_NODESC * 32, B), B, 0, stream>>>(
        xin, bufout, bb, batch, bufin, gsum, gsq, N_NODESC);
    k_norm_relu<<<cdiv((long long)N_NODESC * DIMC, B), B, 0, stream>>>(
        bufin, batch, gsum, gsq, gcnt, lw, lb, N_NODESC * DIMC);
  };

  layer(x,     W1, as1, ad1, b1, lw1, lb1);   // writes activations into bufin
  layer(bufin, W2, as2, ad2, b2, lw2, lb2);   // in-place residual on bufin (elementwise safe)

  k_pool<<<cdiv((long long)N_NODESC * DIMC, B), B, 0, stream>>>(bufin, batch, pooled, N_NODESC * DIMC);
  k_head<<<cdiv(NUM_G * OUT_DIMC, B), B, 0, stream>>>(pooled, gcnt, Wh, bh, out);
}